// BalancedMSELoss_29815662969510
// MI455X (gfx1250) — compile-verified
//
#include <hip/hip_runtime.h>

typedef __attribute__((ext_vector_type(2))) float v2f;
typedef __attribute__((ext_vector_type(8))) float v8f;

#define LOG2E 1.4426950408889634f

__device__ __forceinline__ float fast_exp2(float x) {
#if __has_builtin(__builtin_amdgcn_exp2f)
    return __builtin_amdgcn_exp2f(x);   // raw v_exp_f32 (computes 2^x)
#else
    return exp2f(x);
#endif
}

// Kernel 1: per 16-row stripe, rowsum[i] = sum_j exp(pred_i*tgt_j - 0.5*pred_i^2 - 0.5*tgt_j^2)
// The FULL scaled logit is produced by one V_WMMA_F32_16X16X4_F32 per tile using the K slots:
//   K=0: (log2e*pred_m) * tgt_n
//   K=1:  1.0           * (-0.5*log2e*tgt_n^2)
//   K=2: (-0.5*log2e*pred_m^2) * 1.0
//   K=3:  0 * 0
// so D[m,n] = log2e*(p*t - p^2/2 - t^2/2) and exp2(D) is the Gaussian kernel term.
__global__ __launch_bounds__(256) void bml_rowsum_kernel(
    const float* __restrict__ pred, const float* __restrict__ tgt,
    float* __restrict__ rowsum, int N)
{
    __shared__ float lds[8][16];

    const int tid     = threadIdx.x;
    const int lane    = tid & 31;
    const int wave    = tid >> 5;
    const int rowBase = blockIdx.x * 16;
    const int n       = lane & 15;
    const bool hi     = lane >= 16;

    // A fragment (16x4 f32), loop-invariant.
    // VGPR0: lanes 0-15 -> (M=lane, K=0) = log2e*p ; lanes 16-31 -> (M=lane-16, K=2) = -0.5*log2e*p^2
    // VGPR1: lanes 0-15 -> (M=lane, K=1) = 1.0     ; lanes 16-31 -> (M=lane-16, K=3) = 0
    const float pm = pred[rowBase + n];
    v2f a;
    a.x = hi ? (-0.5f * LOG2E) * pm * pm : LOG2E * pm;
    a.y = hi ? 0.0f : 1.0f;

    float acc[8];
#pragma unroll
    for (int r = 0; r < 8; ++r) acc[r] = 0.0f;

    // Scalar (SGPR-only) loop bound: EXEC stays all-1s around the WMMA.
    const int tilesPerWave = N >> 7;      // (N/16 col tiles) / 8 waves
    int col = wave << 4;                  // this wave's first column; step 8*16 = 128
#pragma unroll 2
    for (int t = 0; t < tilesPerWave; ++t, col += 128) {
        const float tv  = tgt[col + n];
        const float htn = (-0.5f * LOG2E) * tv * tv;

        // B fragment (4x16 f32):
        // VGPR0: lanes 0-15 -> (K=0, N=lane) = t ; lanes 16-31 -> (K=2, N=lane-16) = 1.0
        // VGPR1: lanes 0-15 -> (K=1, N=lane) = -0.5*log2e*t^2 ; lanes 16-31 -> (K=3) = 0
        v2f b;
        b.x = hi ? 1.0f : tv;
        b.y = hi ? 0.0f : htn;

        v8f c = {};
        c = __builtin_amdgcn_wmma_f32_16x16x4_f32(
            /*neg_a=*/false, a, /*neg_b=*/false, b,
            /*c_mod=*/(short)0, c, /*reuse_a=*/false, /*reuse_b=*/false);

#pragma unroll
        for (int r = 0; r < 8; ++r) {
            // scaled logits <= 0 -> exp2 in (0,1], single-pass sum is stable in f32
            acc[r] += fast_exp2(c[r]);
        }
    }

    // Reduce across the 16-lane halves (xor offsets < 16 keep halves independent).
#pragma unroll
    for (int r = 0; r < 8; ++r) {
#pragma unroll
        for (int off = 1; off < 16; off <<= 1)
            acc[r] += __shfl_xor(acc[r], off, 32);
    }

    if (lane == 0) {
#pragma unroll
        for (int r = 0; r < 8; ++r) lds[wave][r] = acc[r];       // rows 0..7 of tile
    }
    if (lane == 16) {
#pragma unroll
        for (int r = 0; r < 8; ++r) lds[wave][8 + r] = acc[r];   // rows 8..15 of tile
    }
    __syncthreads();

    if (tid < 16) {
        float s = 0.0f;
#pragma unroll
        for (int w = 0; w < 8; ++w) s += lds[w][tid];
        rowsum[rowBase + tid] = s;
    }
}

// Kernel 2: loss = (2/N) * sum_i ( log(rowsum_i) + 0.5*(pred_i - tgt_i)^2 )
__global__ __launch_bounds__(256) void bml_finalize_kernel(
    const float* __restrict__ pred, const float* __restrict__ tgt,
    const float* __restrict__ rowsum, float* __restrict__ out, int N)
{
    __shared__ float red[256];
    float s = 0.0f;
    for (int i = threadIdx.x; i < N; i += 256) {
        const float d = pred[i] - tgt[i];
        s += __logf(rowsum[i]) + 0.5f * d * d;
    }
    red[threadIdx.x] = s;
    __syncthreads();
    for (int off = 128; off > 0; off >>= 1) {
        if (threadIdx.x < off) red[threadIdx.x] += red[threadIdx.x + off];
        __syncthreads();
    }
    if (threadIdx.x == 0) out[0] = red[0] * (2.0f / (float)N);
}

extern "C" void kernel_launch(void* const* d_in, const int* in_sizes, int n_in,
                              void* d_out, int out_size, void* d_ws, size_t ws_size,
                              hipStream_t stream) {
    const float* pred = (const float*)d_in[0];  // inputs (N,1) f32; column 0 == flat
    const float* tgt  = (const float*)d_in[1];  // targets (N,) f32
    float* out        = (float*)d_out;          // scalar f32 loss
    float* rowsum     = (float*)d_ws;           // N floats of scratch (64 KB for N=16384)

    const int N = in_sizes[0];
    const int rowTiles = N / 16;                // N = 16384 -> 1024 blocks of 8 waves

    bml_rowsum_kernel<<<rowTiles, 256, 0, stream>>>(pred, tgt, rowsum, N);
    bml_finalize_kernel<<<1, 256, 0, stream>>>(pred, tgt, rowsum, out, N);
}